// MixSizeNumQuatEmbedding_40759239639135
// MI455X (gfx1250) — compile-verified
//
#include <hip/hip_runtime.h>

// MixSizeNumQuatEmbedding for MI455X (gfx1250, wave32).
// Memory-bound gather kernel: one wave32 per output row (b,f); each lane owns
// 4 contiguous floats of the D=128 row.
//
// Roofline: ~0.29 GFLOP vs ~0.67 GB moved -> pure bandwidth/gather bound
// (~30us at 23.3 TB/s). Codebooks (~51MB) + idx tables (~23MB) are
// L2-resident (192MB). Strategy:
//   * 7 codebook gathers per row, each one coalesced global_load_b128/lane.
//   * ALL idx-table reads on the scalar path: idx[(xg)*Mn + 0..Mn-1] is a
//     wave-uniform contiguous run -> s_load_b32/b64/b128 (KMcnt), then
//     per-lane segment selection via v_cndmask. Saves 4 VMEM gathers/row.
//   * Output (82MB stream) written with non-temporal th hint so it doesn't
//     evict the gather tables from L2.

typedef float v4f __attribute__((ext_vector_type(4)));

#define B_SZ      4096
#define F_SZ      39
#define D_SZ      128
#define FIELD_DIM 10000
#define NPAIRS    7

struct Tables {
    const float* cb[NPAIRS];
    const int*   idx[NPAIRS];
};

__global__ __launch_bounds__(256) void MixSizeNumQuatEmbedding_kernel(
    const int*   __restrict__ x,          // (B*F,) int32, values in [0, FIELD_DIM)
    const float* __restrict__ arch_prob,  // (F, 9) fp32
    Tables t,
    float* __restrict__ out,              // (B*F, D) fp32
    int nrows)
{
    // Per-pair constants, folded at compile time by the unrolled loop.
    const int Mn[NPAIRS]  = {1, 1, 2, 4, 1, 2, 4}; // M_SPACE[m] per PAIRS order
    const int col[NPAIRS] = {0, 3, 4, 5, 6, 7, 8}; // j*3 + m column in arch_prob

    const int lane = threadIdx.x & 31;
    // Row index is wave-uniform by construction; force it into an SGPR so the
    // x / arch_prob / idx loads become scalar (s_load) traffic.
    const int row = __builtin_amdgcn_readfirstlane(
        (int)(blockIdx.x * 8u + (threadIdx.x >> 5)));
    if (row >= nrows) return;

    const int f  = row % F_SZ;
    const int xg = __builtin_amdgcn_readfirstlane(x[row] + f * FIELD_DIM);

    const int d0 = lane << 2;   // first of 4 contiguous output elems this lane owns

    v4f acc = {0.f, 0.f, 0.f, 0.f};

#pragma unroll
    for (int p = 0; p < NPAIRS; ++p) {
        const int mn = Mn[p];

        // Segment id for this lane: s = d0 / (128/mn). All 4 elements a lane
        // owns fall inside one segment (plen >= 32, d0 multiple of 4).
        int iv;
        if (mn == 1) {
            // Uniform address -> s_load_b32.
            iv = t.idx[p][xg];
        } else if (mn == 2) {
            // Uniform contiguous pair -> s_load_b64, then lane select.
            const int* ip = t.idx[p] + xg * 2;
            const int i0 = ip[0];
            const int i1 = ip[1];
            const int s  = lane >> 4;          // 0 or 1
            iv = (s != 0) ? i1 : i0;           // one v_cndmask
        } else { // mn == 4
            // Uniform contiguous quad -> s_load_b128, then 2-level select.
            const int* ip = t.idx[p] + xg * 4;
            const int i0 = ip[0];
            const int i1 = ip[1];
            const int i2 = ip[2];
            const int i3 = ip[3];
            const int s  = lane >> 3;          // 0..3
            const int lo = (s & 1) ? i1 : i0;
            const int hi = (s & 1) ? i3 : i2;
            iv = (s & 2) ? hi : lo;            // three v_cndmask
        }

        // Flat codebook offset: iv*D + s*plen + (d0 - s*plen) == iv*D + d0.
        const v4f v    = *(const v4f*)(t.cb[p] + (long)iv * D_SZ + d0);
        const float sc = arch_prob[f * 9 + col[p]];   // uniform -> s_load
        acc += sc * v;                                // 4x v_fmac_f32 per pair
    }

    // Streaming output: non-temporal hint keeps codebooks resident in L2.
    __builtin_nontemporal_store(acc, (v4f*)(out + (long)row * D_SZ + d0));
}

extern "C" void kernel_launch(void* const* d_in, const int* in_sizes, int n_in,
                              void* d_out, int out_size, void* d_ws, size_t ws_size,
                              hipStream_t stream) {
    // Input order (setup_inputs): x, arch_prob, then (cb, idx) for each pair in
    // PAIRS order: (0,0),(1,0),(1,1),(1,2),(2,0),(2,1),(2,2).
    const int*   x         = (const int*)d_in[0];
    const float* arch_prob = (const float*)d_in[1];

    Tables t;
    for (int p = 0; p < NPAIRS; ++p) {
        t.cb[p]  = (const float*)d_in[2 + 2 * p];
        t.idx[p] = (const int*)  d_in[3 + 2 * p];
    }

    float* out = (float*)d_out;
    const int nrows = in_sizes[0];           // B*F = 159744

    const int waves_per_block = 8;           // 256 threads / wave32
    const int grid = (nrows + waves_per_block - 1) / waves_per_block;

    MixSizeNumQuatEmbedding_kernel<<<grid, 256, 0, stream>>>(
        x, arch_prob, t, out, nrows);
}